// MultiAttnVector_5703716569223
// MI455X (gfx1250) — compile-verified
//
#include <hip/hip_runtime.h>
#include <math.h>

#define BB 16
#define NN 4096
#define HH 8
#define DD 64
#define NUM_TYPES 8
#define NUM_SEG 512

typedef __attribute__((ext_vector_type(2))) float v2f;
typedef __attribute__((ext_vector_type(8))) float v8f;

// ---------------------------------------------------------------------------
// Pass 1: attns[b,n,h] = dot(x[b,n,h,:], attn_vector[types[b,n],0,h,:]) / 8
// One wave32 per (b, 16-token tile, head). 16 chained V_WMMA_F32_16X16X4_F32
// ops compute the 16x16 (token x type) dot-product tile; each token keeps the
// column selected by its type. Columns 8-15 of B duplicate columns 0-7 (valid
// finite data, never selected) so ALL loads are unconditional -> no EXEC
// branching, plain global_load_b64 per operand per step.
// ---------------------------------------------------------------------------
__global__ __launch_bounds__(256) void attns_wmma_kernel(
    const float* __restrict__ x,      // [B,N,H,D]
    const float* __restrict__ av,     // [T,1,H,D]
    const int*   __restrict__ types,  // [B,N]
    float*       __restrict__ attns)  // [B,N,H] (workspace)
{
    const int lane = threadIdx.x & 31;
    const int wave = threadIdx.x >> 5;
    const int tile = blockIdx.x * 8 + wave;       // B * (N/16) * H tiles
    const int h    = tile & (HH - 1);
    const int nt   = (tile >> 3) & (NN / 16 - 1);
    const int b    = tile >> 11;
    const int n0   = nt * 16;

    const int half = lane >> 4;   // 0 => K = 4k+{0,1}; 1 => K = 4k+{2,3}
    const int l15  = lane & 15;   // A: row (token-in-tile); B: column (type)

    // A operand source: row = token n0+l15, K base offset 2*half (8B aligned)
    const float* arow = x + (((size_t)((b * NN) + n0 + l15) * HH + h) * DD) + 2 * half;
    // B operand source: column = type (l15 & 7); cols 8-15 duplicate 0-7
    const int tcol = l15 & (NUM_TYPES - 1);
    const float* bcol = av + (((size_t)(tcol * HH) + h) * DD) + 2 * half;

    v8f c = {0.f, 0.f, 0.f, 0.f, 0.f, 0.f, 0.f, 0.f};
#pragma unroll
    for (int kc = 0; kc < DD / 4; ++kc) {
        const v2f a  = *reinterpret_cast<const v2f*>(arow + 4 * kc);
        const v2f bm = *reinterpret_cast<const v2f*>(bcol + 4 * kc);
        // D = A(16x4,f32) * B(4x16,f32) + C   -> v_wmma_f32_16x16x4_f32
        c = __builtin_amdgcn_wmma_f32_16x16x4_f32(
            /*neg_a=*/false, a, /*neg_b=*/false, bm,
            /*c_mod=*/(short)0, c, /*reuse_a=*/false, /*reuse_b=*/false);
    }

    // C/D layout: VGPR r, lanes 0-15 -> M=r, N=lane; lanes 16-31 -> M=r+8.
    // Token m's result against type t lives in column n==t; exactly one lane
    // per row matches and writes (1/sqrt(64) = 0.125 scaling applied here).
#pragma unroll
    for (int r = 0; r < 8; ++r) {
        const int m     = r + 8 * half;
        const int token = n0 + m;
        const int t     = types[b * NN + token];
        if (l15 == t) {
            attns[(size_t)(b * NN + token) * HH + h] = c[r] * 0.125f;
        }
    }
}

// ---------------------------------------------------------------------------
// Pass 2: deterministic segment softmax. indexs is sorted along n, so each
// (b, seg, h) owns a contiguous range found by binary search. One thread per
// (b, seg, h): max -> sum(exp) -> normalize, all in fixed order (no atomics).
// ---------------------------------------------------------------------------
__global__ __launch_bounds__(256) void seg_softmax_kernel(
    const float* __restrict__ attns,   // [B,N,H]
    const int*   __restrict__ indexs,  // [B,N] sorted per row
    float*       __restrict__ out)     // [B,N,H]
{
    const int tid = blockIdx.x * blockDim.x + threadIdx.x;
    if (tid >= BB * NUM_SEG * HH) return;
    const int h   = tid & (HH - 1);
    const int seg = (tid >> 3) & (NUM_SEG - 1);
    const int b   = tid >> 12;

    const int* row = indexs + b * NN;

    // lower_bound(seg)
    int lo = 0, cnt = NN;
    while (cnt > 0) {
        int step = cnt >> 1, mid = lo + step;
        if (row[mid] < seg) { lo = mid + 1; cnt -= step + 1; }
        else                { cnt = step; }
    }
    // upper_bound(seg)
    int hi = lo, cnt2 = NN - lo;
    while (cnt2 > 0) {
        int step = cnt2 >> 1, mid = hi + step;
        if (row[mid] <= seg) { hi = mid + 1; cnt2 -= step + 1; }
        else                 { cnt2 = step; }
    }
    if (hi == lo) return;  // empty segment

    float mx = -INFINITY;
    for (int n = lo; n < hi; ++n)
        mx = fmaxf(mx, attns[(size_t)(b * NN + n) * HH + h]);

    float s = 0.f;
    for (int n = lo; n < hi; ++n)
        s += expf(attns[(size_t)(b * NN + n) * HH + h] - mx);

    const float inv = 1.f / (s + 1e-16f);
    for (int n = lo; n < hi; ++n) {
        const float e = expf(attns[(size_t)(b * NN + n) * HH + h] - mx);
        out[(size_t)(b * NN + n) * HH + h] = e * inv;
    }
}

extern "C" void kernel_launch(void* const* d_in, const int* in_sizes, int n_in,
                              void* d_out, int out_size, void* d_ws, size_t ws_size,
                              hipStream_t stream) {
    const float* x      = (const float*)d_in[0];  // [B,N,H,D] fp32
    const float* av     = (const float*)d_in[1];  // [T,1,H,D] fp32
    const int*   types  = (const int*)d_in[2];    // [B,N] int32
    const int*   indexs = (const int*)d_in[3];    // [B,N] int32 (sorted)
    float*       out    = (float*)d_out;          // [B,N,H] fp32
    float*       attns  = (float*)d_ws;           // [B,N,H] fp32 scratch (2 MB)

    const int tiles = BB * (NN / 16) * HH;        // 32768 wave-tiles
    attns_wmma_kernel<<<tiles / 8, 256, 0, stream>>>(x, av, types, attns);

    const int t2 = BB * NUM_SEG * HH;             // 65536 threads
    seg_softmax_kernel<<<t2 / 256, 256, 0, stream>>>(attns, indexs, out);
}